// ProposalTargetLayer_12412455485710
// MI455X (gfx1250) — compile-verified
//
#include <hip/hip_runtime.h>
#include <cstdint>

// ---------------------------------------------------------------------------
// ProposalTargetLayer: pairwise IoU [16384 x 2048] + global top-10 mask.
//
// Roofline (MI455X): output = 2 * 16384*2048*4B = 268 MB of write-once data
// -> 268MB / 23.3 TB/s ~= 11.5 us store-bound floor. Inputs are tiny (gt
// boxes = 32KB, staged to LDS via the CDNA5 async global->LDS path).
// Each thread owns 4 consecutive columns so both outputs are written with
// b128 non-temporal stores (512B contiguous per wave-store, L2-bypassing).
// ---------------------------------------------------------------------------

typedef float  v4f __attribute__((ext_vector_type(4)));

constexpr int NN    = 16384;   // proposals
constexpr int MM    = 2048;    // gt boxes
constexpr int ROWS  = 16;      // proposal rows per block
constexpr int BT    = 256;     // threads per block (8 waves of 32)
constexpr int VEC   = 4;       // columns per thread per chunk
constexpr int K     = 16;      // per-thread / per-wave top-K list size
constexpr int KEEP  = 10;      // k of top-k
constexpr int NBLK  = NN / ROWS;           // 1024
constexpr int WAVES = BT / 32;             // 8
constexpr int NCAND = NBLK * WAVES * KEEP; // 81920 candidates for pass 2

// ---- sorted-descending register list helpers (all indices compile-time) ----

__device__ __forceinline__ void insertK(float (&v)[K], int (&x)[K], float nv, int nx) {
    if (nv > v[K - 1]) {
        v[K - 1] = nv; x[K - 1] = nx;
#pragma unroll
        for (int k = K - 1; k > 0; --k) {
            bool p = v[k] > v[k - 1];
            float tv = p ? v[k - 1] : v[k];
            float th = p ? v[k]     : v[k - 1];
            int   ti = p ? x[k - 1] : x[k];
            int   tj = p ? x[k]     : x[k - 1];
            v[k - 1] = th; v[k] = tv;
            x[k - 1] = tj; x[k] = ti;
        }
    }
}

// Merge two descending sorted K-lists, keep top-K (FAISS-style bitonic merge).
__device__ __forceinline__ void mergeK(float (&v)[K], int (&x)[K],
                                       const float (&ov)[K], const int (&ox)[K]) {
    float mv[K]; int mx[K];
#pragma unroll
    for (int k = 0; k < K; ++k) {            // elementwise max vs reversed list
        bool p = v[k] >= ov[K - 1 - k];
        mv[k] = p ? v[k] : ov[K - 1 - k];
        mx[k] = p ? x[k] : ox[K - 1 - k];
    }
#pragma unroll
    for (int d = K / 2; d >= 1; d >>= 1) {   // bitonic clean -> descending
#pragma unroll
        for (int k = 0; k < K; ++k) {
            if ((k & d) == 0) {
                int j = k | d;
                bool p = mv[k] >= mv[j];
                float hv = p ? mv[k] : mv[j], lv = p ? mv[j] : mv[k];
                int   hx = p ? mx[k] : mx[j], lx = p ? mx[j] : mx[k];
                mv[k] = hv; mv[j] = lv; mx[k] = hx; mx[j] = lx;
            }
        }
    }
#pragma unroll
    for (int k = 0; k < K; ++k) { v[k] = mv[k]; x[k] = mx[k]; }
}

// wave32 butterfly reduction: afterwards every lane holds the wave top-K.
__device__ __forceinline__ void waveMerge(float (&v)[K], int (&x)[K]) {
#pragma unroll
    for (int m = 1; m < 32; m <<= 1) {
        float ov[K]; int ox[K];
#pragma unroll
        for (int k = 0; k < K; ++k) {
            ov[k] = __shfl_xor(v[k], m, 32);
            ox[k] = __shfl_xor(x[k], m, 32);
        }
        mergeK(v, x, ov, ox);
    }
}

__device__ __forceinline__ float iou_one(const float4 a, const float4 b,
                                         float sumAreaEps) {
    float lx = fmaxf(a.x, b.x), ly = fmaxf(a.y, b.y);
    float rx = fminf(a.z, b.z), ry = fminf(a.w, b.w);
    float w  = fmaxf(rx - lx, 0.0f);
    float h  = fmaxf(ry - ly, 0.0f);
    float inter = w * h;
    float uni   = sumAreaEps - inter;        // areaA + areaB + eps - inter
    return inter * __builtin_amdgcn_rcpf(uni);
}

// ---------------------------------------------------------------------------
// Kernel 1: block = 16 proposal rows x all 2048 cols; each thread owns 4
// consecutive cols per chunk (2 chunks). gt boxes staged via async DMA.
// ---------------------------------------------------------------------------
__global__ __launch_bounds__(BT) void iou_topk_kernel(
    const float4* __restrict__ prop, const float4* __restrict__ gt,
    float* __restrict__ out_iou, float* __restrict__ out_mask,
    float* __restrict__ wsv, int* __restrict__ wsi)
{
    __shared__ float4 sB[MM];     // 32 KB of gt boxes (fits easily in 320KB LDS)
    __shared__ float4 sA[ROWS];
    __shared__ float  sAE[ROWS];  // areaA + eps per row

    const int tx   = threadIdx.x;
    const int row0 = blockIdx.x * ROWS;

    // --- CDNA5 async global->LDS staging of all gt boxes (8 B128 per lane) ---
    unsigned ldsb = (unsigned)(unsigned long long)(&sB[0]); // low 32b = LDS offset
#pragma unroll
    for (int i = 0; i < MM / BT; ++i) {
        int j = i * BT + tx;
        unsigned long long ga = (unsigned long long)(gt + j);
        unsigned dst = ldsb + (unsigned)(j * 16);
        asm volatile("global_load_async_to_lds_b128 %0, %1, off"
                     :: "v"(dst), "v"(ga) : "memory");
    }
    if (tx < ROWS) {
        float4 a = prop[row0 + tx];
        sA[tx]  = a;
        sAE[tx] = (a.z - a.x) * (a.w - a.y) + 1e-8f;
    }
    asm volatile("s_wait_asynccnt 0" ::: "memory");
    __syncthreads();

    float tv[K]; int ti[K];
#pragma unroll
    for (int k = 0; k < K; ++k) { tv[k] = 0.0f; ti[k] = 0; }

#pragma unroll
    for (int c = 0; c < MM; c += BT * VEC) {  // 2 chunks of 1024 cols
        const int col = c + tx * VEC;
        float4 b[VEC]; float areaB[VEC];
#pragma unroll
        for (int u = 0; u < VEC; ++u) {
            b[u]     = sB[col + u];
            areaB[u] = (b[u].z - b[u].x) * (b[u].w - b[u].y);
        }
#pragma unroll
        for (int r = 0; r < ROWS; ++r) {
            const float4 a  = sA[r];              // LDS broadcast (uniform addr)
            const float  ae = sAE[r];
            float io[VEC];
#pragma unroll
            for (int u = 0; u < VEC; ++u)
                io[u] = iou_one(a, b[u], ae + areaB[u]);

            const int idx = (row0 + r) * MM + col;     // 16B aligned
            v4f pack = { io[0], io[1], io[2], io[3] };
            __builtin_nontemporal_store(pack,     (v4f*)(out_iou  + idx));
            __builtin_nontemporal_store((v4f)0.f, (v4f*)(out_mask + idx));

            // rare path: only dive into inserts if any of the 4 can place
            float m = fmaxf(fmaxf(io[0], io[1]), fmaxf(io[2], io[3]));
            if (m > tv[K - 1]) {
#pragma unroll
                for (int u = 0; u < VEC; ++u)
                    insertK(tv, ti, io[u], idx + u);
            }
        }
    }

    waveMerge(tv, ti);
    const int lane = tx & 31, wv = tx >> 5;
    if (lane == 0) {
        int base = (blockIdx.x * WAVES + wv) * KEEP;
#pragma unroll
        for (int k = 0; k < KEEP; ++k) { wsv[base + k] = tv[k]; wsi[base + k] = ti[k]; }
    }
}

// ---------------------------------------------------------------------------
// Kernel 2: one block reduces 81920 (val,idx) candidates to global top-10 and
// scatters ten 1.0f values into the already-zeroed mask half of d_out.
// ---------------------------------------------------------------------------
__global__ __launch_bounds__(BT) void topk_final_kernel(
    const float* __restrict__ wsv, const int* __restrict__ wsi,
    float* __restrict__ out_mask)
{
    __shared__ float lv[WAVES][K];
    __shared__ int   li[WAVES][K];
    const int tx = threadIdx.x;

    float tv[K]; int ti[K];
#pragma unroll
    for (int k = 0; k < K; ++k) { tv[k] = -1.0f; ti[k] = 0; }

    for (int i = tx; i < NCAND; i += BT)
        insertK(tv, ti, wsv[i], wsi[i]);

    waveMerge(tv, ti);
    const int lane = tx & 31, wv = tx >> 5;
    if (lane == 0) {
#pragma unroll
        for (int k = 0; k < K; ++k) { lv[wv][k] = tv[k]; li[wv][k] = ti[k]; }
    }
    __syncthreads();

    if (tx == 0) {
        float fv[K]; int fi[K];
#pragma unroll
        for (int k = 0; k < K; ++k) { fv[k] = lv[0][k]; fi[k] = li[0][k]; }
        for (int w = 1; w < WAVES; ++w)
#pragma unroll
            for (int k = 0; k < K; ++k)
                insertK(fv, fi, lv[w][k], li[w][k]);
#pragma unroll
        for (int k = 0; k < KEEP; ++k)
            out_mask[fi[k]] = 1.0f;
    }
}

// ---------------------------------------------------------------------------
extern "C" void kernel_launch(void* const* d_in, const int* in_sizes, int n_in,
                              void* d_out, int out_size, void* d_ws, size_t ws_size,
                              hipStream_t stream) {
    (void)in_sizes; (void)n_in; (void)out_size; (void)ws_size;
    const float4* prop = (const float4*)d_in[0];  // [N,4] f32
    const float4* gt   = (const float4*)d_in[2];  // [M,4] f32  (cls inputs unused)
    float* out      = (float*)d_out;
    float* out_iou  = out;                                   // [N*M] f32
    float* out_mask = out + (size_t)NN * (size_t)MM;         // [N*M] mask as f32
    float* wsv = (float*)d_ws;                               // 81920 vals
    int*   wsi = (int*)((char*)d_ws + (size_t)NCAND * sizeof(float)); // 81920 idx

    iou_topk_kernel<<<NBLK, BT, 0, stream>>>(prop, gt, out_iou, out_mask, wsv, wsi);
    topk_final_kernel<<<1, BT, 0, stream>>>(wsv, wsi, out_mask);
}